// InfoNCELoss_20633022890108
// MI455X (gfx1250) — compile-verified
//
#include <hip/hip_runtime.h>
#include <math.h>

typedef float v2f __attribute__((ext_vector_type(2)));
typedef float v8f __attribute__((ext_vector_type(8)));

#define B_N 8192
#define D_K 128
#define TILE 64
#define LDS_STRIDE 132   // 132 mod 64 banks = 4 -> conflict-free float2 reads

// workspace layout (float units)
#define WS_TOTAL 0
#define WS_COUNT 1
#define WS_ENORM 16
#define WS_LABF  (WS_ENORM + B_N * D_K)
#define WS_CONFS (WS_LABF + B_N)

__global__ void infonce_init(float* ws) {
    ws[WS_TOTAL] = 0.0f;
    ws[WS_COUNT] = 0.0f;
}

// One wave (32 lanes) per row: L2-normalize + precompute label sign / conf sign.
__global__ __launch_bounds__(256) void infonce_normalize(
    const float* __restrict__ emb,
    const int* __restrict__ labels,
    const float* __restrict__ conf,
    float* __restrict__ ws) {
    const int w    = threadIdx.x >> 5;
    const int lane = threadIdx.x & 31;
    const int row  = blockIdx.x * 8 + w;

    const float4 v = *(const float4*)(emb + row * D_K + lane * 4);
    float s = v.x * v.x + v.y * v.y + v.z * v.z + v.w * v.w;
    #pragma unroll
    for (int off = 1; off < 32; off <<= 1) s += __shfl_xor(s, off, 32);

    const float nrm   = sqrtf(s);
    const float scale = 1.0f / fmaxf(nrm, 1e-12f);
    float4 o;
    o.x = v.x * scale; o.y = v.y * scale; o.z = v.z * scale; o.w = v.w * scale;
    *(float4*)(ws + WS_ENORM + row * D_K + lane * 4) = o;

    if (lane == 0) {
        ws[WS_LABF + row] = (float)(2 * labels[row] - 1);
        const float c = conf[row];
        ws[WS_CONFS + row] = (c > 0.0f) ? 1.0f : ((c < 0.0f) ? -1.0f : 0.0f);
    }
}

// Each block: 64-row band. Loop over 64-col tiles of the Gram matrix.
// 8 waves: wave w -> rows [wr*16, wr*16+16), col half wc*32 (two 16x16 subtiles).
// fp32 WMMA 16x16x4 chained 32x over K=128; masked exp epilogue streamed.
__global__ __launch_bounds__(256) void infonce_main(float* __restrict__ ws) {
    __shared__ float btile[TILE * LDS_STRIDE];
    __shared__ float rowpos[TILE];
    __shared__ float rowneg[TILE];

    const float* __restrict__ e    = ws + WS_ENORM;
    const float* __restrict__ labf = ws + WS_LABF;
    const float* __restrict__ cons = ws + WS_CONFS;

    const int tid   = threadIdx.x;
    const int w     = tid >> 5;
    const int lane  = tid & 31;
    const int wr    = w >> 1;       // row group 0..3
    const int wc    = w & 1;        // column half 0..1
    const int m16   = lane & 15;
    const int hi    = lane >> 4;    // 0 or 1
    const int khalf = hi * 2;       // A/B fp32 WMMA K-pair selector
    const int i0    = blockIdx.x * TILE;

    if (tid < TILE) { rowpos[tid] = 0.0f; rowneg[tid] = 0.0f; }

    // Preload A fragments (16x4 fp32 layout): row fixed per lane, 32 K-steps.
    const int arow = i0 + wr * 16 + m16;
    v2f a_frag[32];
    #pragma unroll
    for (int kk = 0; kk < 32; ++kk)
        a_frag[kk] = *(const v2f*)(e + arow * D_K + 4 * kk + khalf);

    // Row tags for the 8 C/D slots this lane holds (rows i0 + wr*16 + 8*hi + r).
    float ilab[8], icon[8];
    #pragma unroll
    for (int r = 0; r < 8; ++r) {
        const int gi = i0 + wr * 16 + 8 * hi + r;
        ilab[r] = labf[gi];
        icon[r] = cons[gi];
    }

    float posA[8], negA[8];
    #pragma unroll
    for (int r = 0; r < 8; ++r) { posA[r] = 0.0f; negA[r] = 0.0f; }

    for (int jt = 0; jt < B_N / TILE; ++jt) {
        const int j0 = jt * TILE;
        __syncthreads();
        // Stage 64x128 column tile into LDS (8 float4 per thread).
        #pragma unroll
        for (int it = 0; it < 8; ++it) {
            const int idx  = tid + it * 256;     // 0..2047
            const int rrow = idx >> 5;           // 0..63
            const int c4   = (idx & 31) * 4;     // 0..124
            const float4 v = *(const float4*)(e + (j0 + rrow) * D_K + c4);
            *(float4*)(&btile[rrow * LDS_STRIDE + c4]) = v;
        }
        __syncthreads();

        const int gj0 = j0 + wc * 32 + m16;   // subtile-0 column of this lane
        const int gj1 = gj0 + 16;             // subtile-1 column
        const float jl0 = labf[gj0], jc0 = cons[gj0];
        const float jl1 = labf[gj1], jc1 = cons[gj1];

        v8f acc0 = {};
        v8f acc1 = {};
        const int lr0 = wc * 32 + m16;        // local LDS row (== column index n)
        const int lr1 = lr0 + 16;
        #pragma unroll
        for (int kk = 0; kk < 32; ++kk) {
            const v2f a  = a_frag[kk];
            const v2f b0 = *(const v2f*)(&btile[lr0 * LDS_STRIDE + 4 * kk + khalf]);
            const v2f b1 = *(const v2f*)(&btile[lr1 * LDS_STRIDE + 4 * kk + khalf]);
            acc0 = __builtin_amdgcn_wmma_f32_16x16x4_f32(
                false, a, false, b0, (short)0, acc0, false, false);
            acc1 = __builtin_amdgcn_wmma_f32_16x16x4_f32(
                false, a, false, b1, (short)0, acc1, false, false);
        }

        // Masked exp epilogue; branchless so EXEC stays all-ones around WMMA.
        #pragma unroll
        for (int r = 0; r < 8; ++r) {
            const int gi = i0 + wr * 16 + 8 * hi + r;
            const float ex0 = expf(acc0[r] * 10.0f);   // /TEMPERATURE
            const float ex1 = expf(acc1[r] * 10.0f);
            const float lp0 = ilab[r] * jl0, cp0 = icon[r] * jc0;
            const float lp1 = ilab[r] * jl1, cp1 = icon[r] * jc1;
            posA[r] += ((cp0 > 0.0f) && (lp0 > 0.0f) && (gi != gj0)) ? ex0 : 0.0f;
            negA[r] += ((cp0 > 0.0f) && (lp0 < 0.0f)) ? ex0 : 0.0f;
            posA[r] += ((cp1 > 0.0f) && (lp1 > 0.0f) && (gi != gj1)) ? ex1 : 0.0f;
            negA[r] += ((cp1 > 0.0f) && (lp1 < 0.0f)) ? ex1 : 0.0f;
        }
    }

    // Reduce across the 16 lanes sharing each row (xor 1,2,4,8 stays in-half).
    #pragma unroll
    for (int r = 0; r < 8; ++r) {
        float p = posA[r], n = negA[r];
        #pragma unroll
        for (int off = 1; off < 16; off <<= 1) {
            p += __shfl_xor(p, off, 32);
            n += __shfl_xor(n, off, 32);
        }
        posA[r] = p; negA[r] = n;
    }
    if (m16 == 0) {
        const int base = wr * 16 + 8 * hi;
        #pragma unroll
        for (int r = 0; r < 8; ++r) {
            atomicAdd(&rowpos[base + r], posA[r]);   // combine the two col halves
            atomicAdd(&rowneg[base + r], negA[r]);
        }
    }
    __syncthreads();

    if (tid < TILE) {
        const float p = rowpos[tid];
        const float n = rowneg[tid];
        if ((p > 0.0f) && (n > 0.0f)) {
            const float loss = -logf(p / (p + n + 1e-8f));
            atomicAdd(&ws[WS_TOTAL], loss);
            atomicAdd(&ws[WS_COUNT], 1.0f);
        }
    }
}

__global__ void infonce_finalize(const float* __restrict__ ws, float* __restrict__ out) {
    const float t = ws[WS_TOTAL];
    const float c = ws[WS_COUNT];
    out[0] = (c > 0.0f) ? (t / fmaxf(c, 1.0f)) : 0.0f;
}

extern "C" void kernel_launch(void* const* d_in, const int* in_sizes, int n_in,
                              void* d_out, int out_size, void* d_ws, size_t ws_size,
                              hipStream_t stream) {
    (void)in_sizes; (void)n_in; (void)out_size; (void)ws_size;
    const float* emb    = (const float*)d_in[0];
    const int*   labels = (const int*)d_in[1];
    const float* conf   = (const float*)d_in[2];
    float* out = (float*)d_out;
    float* ws  = (float*)d_ws;

    infonce_init<<<1, 1, 0, stream>>>(ws);
    infonce_normalize<<<B_N / 8, 256, 0, stream>>>(emb, labels, conf, ws);
    infonce_main<<<B_N / TILE, 256, 0, stream>>>(ws);
    infonce_finalize<<<1, 1, 0, stream>>>(ws, out);
}